// DualEncoderModel_44083544326601
// MI455X (gfx1250) — compile-verified
//
#include <hip/hip_runtime.h>
#include <hip/hip_bf16.h>
#include <cstdint>

// ---------------- problem constants (from reference) ----------------
#define NN      12000      // DRUG_NUM + TARGET_NUM
#define DRUGS   4000
#define TGTS    8000
#define INDIM   256
#define HG      128
#define HID     256
#define NNZ_G   768000
#define NNZ_DP  100000

typedef __attribute__((ext_vector_type(2))) float        v2f;
typedef __attribute__((ext_vector_type(8))) float        v8f;
typedef __attribute__((ext_vector_type(4))) unsigned int v4u;
typedef __attribute__((ext_vector_type(4))) int          v4i;
typedef __attribute__((ext_vector_type(8))) int          v8i;

enum { ACT_NONE = 0, ACT_LEAKY = 1, ACT_RELU = 2 };

// --------------------------------------------------------------------
// fp32 WMMA: D(16x16) = A(16x4) * B(4x16) + C  (wave32, ISA 7.12.2)
// --------------------------------------------------------------------
__device__ __forceinline__ v8f wmma_f32_16x16x4(v2f a, v2f b, v8f c) {
  return __builtin_amdgcn_wmma_f32_16x16x4_f32(
      /*neg_a=*/false, a, /*neg_b=*/false, b,
      /*c_mod=*/(short)0, c, /*reuse_a=*/false, /*reuse_b=*/false);
}

// --------------------------------------------------------------------
// TDM: stage nElems contiguous f32 from global into LDS at ldsOff.
// 1-D descriptor: data_size=4B, tile_dim0=nElems, tile_dim1=1.
// D# group0: [1:0]=count=1, [63:32]=lds_addr, [120:64]=global_addr,
//            [127:126]=type=2.  (CDNA5 ISA)
// 6-arg builtin form (clang-23 / therock-10.0 toolchain).
// --------------------------------------------------------------------
__device__ __forceinline__ void tdm_load_to_lds(const float* gsrc,
                                                unsigned ldsOff, int nElems)
{
  unsigned long long ga = (unsigned long long)(size_t)gsrc;
  v4u g0;
  g0.x = 1u;                                      // count = 1 (user D#)
  g0.y = ldsOff;                                  // lds_addr (bytes)
  g0.z = (unsigned)(ga & 0xffffffffull);          // global_addr[31:0]
  g0.w = (unsigned)((ga >> 32) & 0x01ffffffull)   // global_addr[56:32]
       | 0x80000000u;                             // type = 2 ("image")
  v8i g1;
  g1[0] = 0x00020000;                             // wg_mask=0, data_size=4B
  g1[1] = (nElems & 0xffff) << 16;                // tensor_dim0[15:0]
  g1[2] = ((nElems >> 16) & 0xffff) | (1 << 16);  // tensor_dim0[31:16] | tensor_dim1=1
  g1[3] = (nElems & 0xffff) << 16;                // tile_dim0 = nElems
  g1[4] = 1;                                      // tile_dim1=1, tile_dim2=0
  g1[5] = nElems;                                 // tensor_dim0_stride[31:0]
  g1[6] = 0;
  g1[7] = 0;
  v4i z4 = {0, 0, 0, 0};                          // groups 2/3 unused (<=2D)
  v8i z8 = {0, 0, 0, 0, 0, 0, 0, 0};              // extra group slot unused
  __builtin_amdgcn_tensor_load_to_lds(g0, g1, z4, z4, z8, 0);
}

// --------------------------------------------------------------------
// Generic C(MxN) = act(A(MxK) @ B(KxN) + bias), one 16x16 tile / wave.
// --------------------------------------------------------------------
__global__ __launch_bounds__(256) void gemm_wmma_f32(
    const float* __restrict__ A, const float* __restrict__ B,
    const float* __restrict__ bias, float* __restrict__ C,
    int M, int N, int K, int act)
{
  const int wave   = (blockIdx.x * blockDim.x + threadIdx.x) >> 5;
  const int lane   = threadIdx.x & 31;
  const int mtiles = M >> 4;
  const int ntiles = N >> 4;
  if (wave >= mtiles * ntiles) return;   // wave-uniform: EXEC stays all-ones

  const int m0   = (wave / ntiles) << 4;
  const int n0   = (wave % ntiles) << 4;
  const int half = lane >> 4;
  const int l    = lane & 15;

  v8f acc = {};
  const float* ap = A + (size_t)(m0 + l) * K + 2 * half;
  const float* bp = B + (size_t)(2 * half) * N + (n0 + l);

  for (int k0 = 0; k0 < K; k0 += 4) {
    v2f a, b;
    a.x = ap[0];
    a.y = ap[1];
    b.x = bp[0];
    b.y = bp[N];
    __builtin_prefetch(ap + 4, 0, 1);
    acc = wmma_f32_16x16x4(a, b, acc);
    ap += 4;
    bp += 4 * N;
  }

  const int   col = n0 + l;
  const float bv  = bias ? bias[col] : 0.0f;
  float* cp = C + (size_t)(m0 + 8 * half) * N + col;
#pragma unroll
  for (int v = 0; v < 8; ++v) {
    float x = acc[v] + bv;
    if (act == ACT_LEAKY)      x = (x > 0.0f) ? x : 0.25f * x;
    else if (act == ACT_RELU)  x = (x > 0.0f) ? x : 0.0f;
    cp[(size_t)v * N] = x;
  }
}

// --------------------------------------------------------------------
// Dominant GEMM: C(MxN) = A(MxK=128) @ B^T, B stored NxK row-major.
// Wave computes a 16x64 strip (4 accumulators); A strip (16x128 = 8 KB,
// contiguous rows) staged per-wave into LDS via TDM; A frags via ds_load;
// output written with nontemporal stores (write-once 128 MB).
// --------------------------------------------------------------------
__global__ __launch_bounds__(256) void gemm_bt_wmma_tdm(
    const float* __restrict__ A, const float* __restrict__ B,
    float* __restrict__ C, int M, int N)   // K == HG == 128
{
  __shared__ float ldsA[8 * 16 * HG];      // 8 waves * 8 KB = 64 KB
  const int wave  = (blockIdx.x * blockDim.x + threadIdx.x) >> 5;
  const int winb  = (threadIdx.x >> 5) & 7;
  const int lane  = threadIdx.x & 31;
  const int mtiles = M >> 4;
  const int nblks  = N >> 6;               // 64-col blocks
  if (wave >= mtiles * nblks) return;      // wave-uniform

  const int m0   = (wave / nblks) << 4;
  const int n0   = (wave % nblks) << 6;
  const int half = lane >> 4;
  const int l    = lane & 15;

  // ---- async TDM stage of this wave's 16-row A strip, then wait ----
  float* myLds = &ldsA[winb * 16 * HG];
  tdm_load_to_lds(A + (size_t)m0 * HG, (unsigned)(size_t)myLds, 16 * HG);
  __builtin_amdgcn_s_wait_tensorcnt(0);

  v8f acc0 = {}, acc1 = {}, acc2 = {}, acc3 = {};
  const float* alds = myLds + l * HG + 2 * half;
  const float* bp0  = B + (size_t)(n0 +  0 + l) * HG + 2 * half;
  const float* bp1  = B + (size_t)(n0 + 16 + l) * HG + 2 * half;
  const float* bp2  = B + (size_t)(n0 + 32 + l) * HG + 2 * half;
  const float* bp3  = B + (size_t)(n0 + 48 + l) * HG + 2 * half;

  for (int k0 = 0; k0 < HG; k0 += 4) {
    v2f a = *(const v2f*)(alds + k0);      // ds_load_b64 from LDS
    v2f b0, b1, b2, b3;
    b0.x = bp0[k0]; b0.y = bp0[k0 + 1];
    b1.x = bp1[k0]; b1.y = bp1[k0 + 1];
    b2.x = bp2[k0]; b2.y = bp2[k0 + 1];
    b3.x = bp3[k0]; b3.y = bp3[k0 + 1];
    __builtin_prefetch(bp0 + k0 + 4, 0, 1);
    acc0 = wmma_f32_16x16x4(a, b0, acc0);
    acc1 = wmma_f32_16x16x4(a, b1, acc1);
    acc2 = wmma_f32_16x16x4(a, b2, acc2);
    acc3 = wmma_f32_16x16x4(a, b3, acc3);
  }

  float* cp = C + (size_t)(m0 + 8 * half) * N + (n0 + l);
#pragma unroll
  for (int v = 0; v < 8; ++v) {
    __builtin_nontemporal_store(acc0[v], cp + (size_t)v * N +  0);
    __builtin_nontemporal_store(acc1[v], cp + (size_t)v * N + 16);
    __builtin_nontemporal_store(acc2[v], cp + (size_t)v * N + 32);
    __builtin_nontemporal_store(acc3[v], cp + (size_t)v * N + 48);
  }
}

// ------------------- SpMM (COO scatter, 128 cols) -------------------
__global__ __launch_bounds__(256) void spmm_scatter(
    const int* __restrict__ row, const int* __restrict__ col,
    const float* __restrict__ val, const float* __restrict__ X,
    float* __restrict__ Y, int nnz)
{
  const int wave = (blockIdx.x * blockDim.x + threadIdx.x) >> 5;
  if (wave >= nnz) return;
  const int lane = threadIdx.x & 31;
  const int r = row[wave];
  const int c = col[wave];
  const float v = val[wave];
  const float4 xv = ((const float4*)(X + (size_t)c * HG))[lane];
  float* yp = Y + (size_t)r * HG + lane * 4;
  atomicAdd(yp + 0, v * xv.x);
  atomicAdd(yp + 1, v * xv.y);
  atomicAdd(yp + 2, v * xv.z);
  atomicAdd(yp + 3, v * xv.w);
}

// even_prop scatter: Y[dst] += dinv[src]*dinv[dst] * X[src]
__global__ __launch_bounds__(256) void prop_scatter(
    const int* __restrict__ src, const int* __restrict__ dst,
    const float* __restrict__ dinv, const float* __restrict__ X,
    float* __restrict__ Y, int nnz)
{
  const int wave = (blockIdx.x * blockDim.x + threadIdx.x) >> 5;
  if (wave >= nnz) return;
  const int lane = threadIdx.x & 31;
  const int s = src[wave];
  const int d = dst[wave];
  const float nrm = dinv[s] * dinv[d];
  const float4 xv = ((const float4*)(X + (size_t)s * HG))[lane];
  float* yp = Y + (size_t)d * HG + lane * 4;
  atomicAdd(yp + 0, nrm * xv.x);
  atomicAdd(yp + 1, nrm * xv.y);
  atomicAdd(yp + 2, nrm * xv.z);
  atomicAdd(yp + 3, nrm * xv.w);
}

// ------------------------- small kernels ----------------------------
__global__ __launch_bounds__(256) void rows_bias_init(
    float* __restrict__ out, const float* __restrict__ bias, int n)
{
  int i = blockIdx.x * blockDim.x + threadIdx.x;
  if (i < n) out[i] = bias[i & (HG - 1)];
}

__global__ __launch_bounds__(256) void act_inplace(
    float* __restrict__ x, int n, float slope)
{
  int i = blockIdx.x * blockDim.x + threadIdx.x;
  if (i < n) {
    float v = x[i];
    x[i] = (v > 0.0f) ? v : slope * v;
  }
}

__global__ __launch_bounds__(256) void deg_count(
    const int* __restrict__ src, float* __restrict__ deg, int nnz)
{
  int i = blockIdx.x * blockDim.x + threadIdx.x;
  if (i < nnz) atomicAdd(&deg[src[i]], 1.0f);
}

__global__ __launch_bounds__(256) void dinv_kernel(
    const float* __restrict__ deg, float* __restrict__ dinv, int n)
{
  int i = blockIdx.x * blockDim.x + threadIdx.x;
  if (i < n) {
    float d = deg[i];
    dinv[i] = (d > 0.0f) ? rsqrtf(d) : 0.0f;
  }
}

// Hc = w*H1 + (1-w)*(alpha*x + alpha*(1-alpha)^2 * p2)
__global__ __launch_bounds__(256) void combine_kernel(
    const float* __restrict__ H1, const float* __restrict__ XS,
    const float* __restrict__ P2, const float* __restrict__ wp,
    float* __restrict__ HC, int n)
{
  int i = blockIdx.x * blockDim.x + threadIdx.x;
  if (i < n) {
    float w = wp[0];
    float h2 = 0.1f * XS[i] + 0.081f * P2[i];   // temp[0], temp[1]
    HC[i] = w * H1[i] + (1.0f - w) * h2;
  }
}

// ------------------------- host-side glue ---------------------------
static inline void launch_gemm(const float* A, const float* B,
                               const float* bias, float* C,
                               int M, int N, int K, int act, hipStream_t s)
{
  int tiles  = (M >> 4) * (N >> 4);
  int blocks = (tiles + 7) / 8;         // 8 waves (256 thr) per block
  gemm_wmma_f32<<<blocks, 256, 0, s>>>(A, B, bias, C, M, N, K, act);
}

static inline int ewb(int n) { return (n + 255) / 256; }

extern "C" void kernel_launch(void* const* d_in, const int* in_sizes, int n_in,
                              void* d_out, int out_size, void* d_ws, size_t ws_size,
                              hipStream_t stream)
{
  const float* H      = (const float*)d_in[0];
  const int*   g_row  = (const int*)d_in[1];
  const int*   g_col  = (const int*)d_in[2];
  const float* g_val  = (const float*)d_in[3];
  const int*   e_src  = (const int*)d_in[4];
  const int*   e_dst  = (const int*)d_in[5];
  const float* W_gc1  = (const float*)d_in[6];
  const float* b_gc1  = (const float*)d_in[7];
  const float* W_gc2  = (const float*)d_in[8];
  const float* b_gc2  = (const float*)d_in[9];
  const float* W_lin1 = (const float*)d_in[10];
  const float* b_lin1 = (const float*)d_in[11];
  const float* W_lin2 = (const float*)d_in[12];
  const float* b_lin2 = (const float*)d_in[13];
  const float* trainW = (const float*)d_in[14];
  const float* wp     = (const float*)d_in[15];
  float* out = (float*)d_out;

  // workspace layout (~29.4 MB, buffers reused across phases)
  float* Abuf = (float*)d_ws;                  // NN*HG
  float* Bbuf = Abuf + (size_t)NN * HG;        // NN*HG
  float* Cbuf = Bbuf + (size_t)NN * HG;        // NN*HG
  float* Dbuf = Cbuf + (size_t)NN * HG;        // NN*HID
  float* DEG  = Dbuf + (size_t)NN * HID;       // NN
  float* DINV = DEG + NN;                      // NN

  const int nHG = NN * HG;
  const int spmm_blocks = (NNZ_G * 32 + 255) / 256;
  const int prop_blocks = (NNZ_DP * 32 + 255) / 256;

  // ---- GCN layer 1: h = leaky(spmm(G, H @ W_gc1) + b_gc1) ----
  launch_gemm(H, W_gc1, nullptr, Abuf, NN, HG, INDIM, ACT_NONE, stream);
  rows_bias_init<<<ewb(nHG), 256, 0, stream>>>(Bbuf, b_gc1, nHG);
  spmm_scatter<<<spmm_blocks, 256, 0, stream>>>(g_row, g_col, g_val, Abuf, Bbuf, NNZ_G);
  act_inplace<<<ewb(nHG), 256, 0, stream>>>(Bbuf, nHG, 0.25f);

  // ---- GCN layer 2: H1 = leaky(spmm(G, h @ W_gc2) + b_gc2) ----
  launch_gemm(Bbuf, W_gc2, nullptr, Abuf, NN, HG, HG, ACT_NONE, stream);
  rows_bias_init<<<ewb(nHG), 256, 0, stream>>>(Cbuf, b_gc2, nHG);
  spmm_scatter<<<spmm_blocks, 256, 0, stream>>>(g_row, g_col, g_val, Abuf, Cbuf, NNZ_G);
  act_inplace<<<ewb(nHG), 256, 0, stream>>>(Cbuf, nHG, 0.25f);      // Cbuf = H1

  // ---- MLP: x = relu(H1@W_lin1+b1) @ W_lin2 + b2 ----
  launch_gemm(Cbuf, W_lin1, b_lin1, Dbuf, NN, HID, HG, ACT_RELU, stream);
  launch_gemm(Dbuf, W_lin2, b_lin2, Abuf, NN, HG, HID, ACT_NONE, stream); // Abuf = x

  // ---- even_prop ----
  (void)hipMemsetAsync(DEG, 0, (size_t)NN * sizeof(float), stream);
  deg_count<<<ewb(NNZ_DP), 256, 0, stream>>>(e_src, DEG, NNZ_DP);
  dinv_kernel<<<ewb(NN), 256, 0, stream>>>(DEG, DINV, NN);
  (void)hipMemsetAsync(Bbuf, 0, (size_t)nHG * sizeof(float), stream);
  prop_scatter<<<prop_blocks, 256, 0, stream>>>(e_src, e_dst, DINV, Abuf, Bbuf, NNZ_DP); // P1
  (void)hipMemsetAsync(Dbuf, 0, (size_t)nHG * sizeof(float), stream);
  prop_scatter<<<prop_blocks, 256, 0, stream>>>(e_src, e_dst, DINV, Bbuf, Dbuf, NNZ_DP); // P2

  // ---- Hc = w*H1 + (1-w)*H2 ----
  combine_kernel<<<ewb(nHG), 256, 0, stream>>>(Cbuf, Abuf, Dbuf, wp, Bbuf, nHG); // Bbuf = Hc

  // ---- output: (HR @ train_W) @ HD^T ----
  launch_gemm(Bbuf, trainW, nullptr, Abuf, DRUGS, HG, HG, ACT_NONE, stream); // T = HR@W
  {
    int tiles  = (DRUGS >> 4) * (TGTS >> 6);   // 250 * 125 = 31250 waves
    int blocks = (tiles + 7) / 8;
    gemm_bt_wmma_tdm<<<blocks, 256, 0, stream>>>(
        Abuf, Bbuf + (size_t)DRUGS * HG, out, DRUGS, TGTS);
  }
}